// ModelNew_1580547971750
// MI455X (gfx1250) — compile-verified
//
#include <hip/hip_runtime.h>
#include <math.h>

// Problem constants (from reference): N=32, CIN=64, COUT=128, K=4, H=W=64
#define NB   32
#define CIN  64
#define COUT 128
#define HW   4096            // 64*64
#define KK   16              // 4*4
#define OUT_HW_INV (1.0f / 4489.0f)   // 1/(67*67)

typedef __attribute__((ext_vector_type(2))) float v2f;
typedef __attribute__((ext_vector_type(8))) float v8f;

// ---------------------------------------------------------------------------
// Kernel 1: Sx[n,ci] = sum over 64x64 spatial plane of x.  (the 32 MB stream)
// One block per (n,ci) plane: 4096 floats = 1024 float4; 256 threads x 4 b128
// loads each.  Pure HBM-bandwidth-bound.
// ---------------------------------------------------------------------------
__global__ __launch_bounds__(256) void reduce_x_kernel(const float* __restrict__ x,
                                                       float* __restrict__ Sx) {
  const int plane = blockIdx.x;                       // 0 .. NB*CIN-1
  const float4* p = (const float4*)(x + (size_t)plane * HW);
  const int tid = threadIdx.x;

  float s = 0.0f;
#pragma unroll
  for (int i = 0; i < 4; ++i) {                       // 4 * 256 = 1024 float4
    float4 v = p[tid + i * 256];
    s += (v.x + v.y) + (v.z + v.w);
  }

  __shared__ float red[256];
  red[tid] = s;
  __syncthreads();
#pragma unroll
  for (int off = 128; off > 0; off >>= 1) {
    if (tid < off) red[tid] += red[tid + off];
    __syncthreads();
  }
  if (tid == 0) Sx[plane] = red[0];
}

// ---------------------------------------------------------------------------
// Kernel 2: Sw[ci,co] = sum over 4x4 kernel taps of weight (CIN,COUT,K,K).
// KK=16 contiguous floats per output -> 4 x float4 per thread.
// Sw stored row-major [CIN][COUT], exactly the B layout the GEMM wants.
// ---------------------------------------------------------------------------
__global__ __launch_bounds__(256) void reduce_w_kernel(const float* __restrict__ w,
                                                       float* __restrict__ Sw) {
  const int idx = blockIdx.x * 256 + threadIdx.x;     // 0 .. CIN*COUT-1
  if (idx < CIN * COUT) {
    const float4* p = (const float4*)(w + (size_t)idx * KK);
    float s = 0.0f;
#pragma unroll
    for (int i = 0; i < 4; ++i) {
      float4 v = p[i];
      s += (v.x + v.y) + (v.z + v.w);
    }
    Sw[idx] = s;
  }
}

// ---------------------------------------------------------------------------
// Kernel 3: pooled[32][128] = Sx(32x64) @ Sw(64x128) * 1/4489 + cb + eb,
// then out[n] = 10 * logsumexp(pooled[n,:]).
// GEMM done with V_WMMA_F32_16X16X4_F32: 16 waves (512 thr), one 16x16 tile
// per wave, K=64 swept in 16 steps of 4.
//
// Fragment layouts per CDNA5 ISA 7.12.2 (wave32):
//   A 16x4 : lane<16 -> row=lane,    {K=0,K=1}; lane>=16 -> row=lane-16, {K=2,K=3}
//   B 4x16 : VGPR0 = K {0 | 2}, VGPR1 = K {1 | 3}; col = lane & 15
//   C 16x16: VGPR j -> M = j + 8*(lane>=16), N = lane & 15
// EXEC is all-1s for every WMMA (no divergence before the barrier).
// ---------------------------------------------------------------------------
__global__ __launch_bounds__(512) void gemm_lse_kernel(const float* __restrict__ Sx,
                                                       const float* __restrict__ Sw,
                                                       const float* __restrict__ cb,
                                                       const float* __restrict__ eb,
                                                       float* __restrict__ out) {
  __shared__ float pooled[NB][COUT];

  const int tid  = threadIdx.x;
  const int lane = tid & 31;
  const int wave = tid >> 5;        // 0..15
  const int mt   = wave >> 3;       // 0..1  (row tile)
  const int nt   = wave & 7;        // 0..7  (col tile)
  const int half = lane >> 4;       // 0/1   (lane half)
  const int l16  = lane & 15;

  const int row = mt * 16 + l16;    // A-matrix row this lane loads
  const int col = nt * 16 + l16;    // B/C column this lane owns

  v8f c = {};
#pragma unroll
  for (int kt = 0; kt < CIN / 4; ++kt) {
    const int kb = kt * 4 + half * 2;     // lane-half's K base within step
    v2f a, b;
    a.x = Sx[row * CIN + kb + 0];
    a.y = Sx[row * CIN + kb + 1];
    b.x = Sw[(kb + 0) * COUT + col];
    b.y = Sw[(kb + 1) * COUT + col];
    // 8 args: (neg_a, A, neg_b, B, c_mod, C, reuse_a, reuse_b)
    c = __builtin_amdgcn_wmma_f32_16x16x4_f32(
        /*neg_a=*/false, a, /*neg_b=*/false, b,
        /*c_mod=*/(short)0, c, /*reuse_a=*/false, /*reuse_b=*/false);
  }

  const float bias = cb[col] + eb[col];
#pragma unroll
  for (int j = 0; j < 8; ++j) {
    const int m = mt * 16 + half * 8 + j;
    pooled[m][col] = c[j] * OUT_HW_INV + bias;
  }
  __syncthreads();

  if (tid < NB) {
    float mx = -INFINITY;
#pragma unroll 4
    for (int i = 0; i < COUT; ++i) mx = fmaxf(mx, pooled[tid][i]);
    float s = 0.0f;
#pragma unroll 4
    for (int i = 0; i < COUT; ++i) s += __expf(pooled[tid][i] - mx);
    out[tid] = 10.0f * (mx + __logf(s));
  }
}

// ---------------------------------------------------------------------------
extern "C" void kernel_launch(void* const* d_in, const int* in_sizes, int n_in,
                              void* d_out, int out_size, void* d_ws, size_t ws_size,
                              hipStream_t stream) {
  const float* x  = (const float*)d_in[0];   // (32,64,64,64)
  const float* w  = (const float*)d_in[1];   // (64,128,4,4)
  const float* cb = (const float*)d_in[2];   // (128,)
  const float* eb = (const float*)d_in[3];   // (128,)
  float* out = (float*)d_out;                // (32,)

  float* Sx = (float*)d_ws;                         // 32*64   floats ( 8 KB)
  float* Sw = Sx + NB * CIN;                        // 64*128  floats (32 KB)

  reduce_x_kernel<<<NB * CIN, 256, 0, stream>>>(x, Sx);
  reduce_w_kernel<<<(CIN * COUT + 255) / 256, 256, 0, stream>>>(w, Sw);
  gemm_lse_kernel<<<1, 512, 0, stream>>>(Sx, Sw, cb, eb, out);
}